// SpatialSmoothingFeature_48825188221610
// MI455X (gfx1250) — compile-verified
//
#include <hip/hip_runtime.h>
#include <math.h>

typedef __attribute__((ext_vector_type(2))) float v2f;
typedef __attribute__((ext_vector_type(4))) float v4f;
typedef __attribute__((ext_vector_type(8))) float v8f;

#define N_BATCH 256
#define M_ROWS  8
#define T_LEN   16384
#define SPLITS  32
#define CHUNK   (T_LEN / SPLITS)      // 512 t-samples per wave
#define PAIRS   (N_BATCH / 2)         // 128 pairs of batches
#define WAVES_PER_BLOCK 8             // 256 threads = 8 wave32

// ---------------------------------------------------------------------------
// Kernel 0: zero the Gram-accumulator workspace (256 * 64 floats).
// ---------------------------------------------------------------------------
__global__ void ssf_zero_ws(float* __restrict__ ws, int n) {
    int i = blockIdx.x * blockDim.x + threadIdx.x;
    if (i < n) ws[i] = 0.0f;
}

// ---------------------------------------------------------------------------
// Kernel 1: streaming Gram matrices via V_WMMA_F32_16X16X4_F32.
//
// Each wave owns (pair, split): a 16-row tile (two batches of 8 rows) and a
// 512-sample slice of T. Per iteration, each lane issues one 16B load
// (global_load_b128) and we fold 8 t-samples with two WMMAs. For this shape
// the A-register layout of rows of X equals the B-register layout of X^T, so
// a single register feeds both operands.
//
// A/B layout (32-bit, 16x4): lanes 0-15 hold row m, K={0,1}; lanes 16-31 hold
// row m, K={2,3}. Low lanes load t..t+3, high lanes t+4..t+7; WMMA#1 uses
// vector elems (0,1) -> covers {t,t+1, t+4,t+5}; WMMA#2 uses (2,3) ->
// {t+2,t+3, t+6,t+7}. Assignment of K-slot to t is consistent between A and B
// (same register), so each of the 8 samples is accumulated exactly once.
// ---------------------------------------------------------------------------
__global__ __launch_bounds__(256) void ssf_gram_wmma(
    const float* __restrict__ X, float* __restrict__ ws) {

    const int wave  = (blockIdx.x * blockDim.x + threadIdx.x) >> 5;
    const int lane  = threadIdx.x & 31;
    const int pair  = wave >> 5;          // / SPLITS
    const int split = wave & (SPLITS - 1);

    const int r16 = lane & 15;            // row within the 16-row tile
    const int hi  = lane >> 4;            // 0: K={0,1}, 1: K={2,3}

    // Global row: tile rows 0..15 map to rows pair*16 .. pair*16+15 of the
    // flattened (256*8, T) matrix (rows 0-7 = batch 2p, rows 8-15 = 2p+1).
    const int grow = pair * 16 + r16;
    const float* rowp = X + (size_t)grow * T_LEN + split * CHUNK + 4 * hi;

    v8f acc = {};
    #pragma unroll 4
    for (int it = 0; it < CHUNK / 8; ++it) {
        v4f d = *(const v4f*)rowp;        // global_load_b128, 16B aligned
        rowp += 8;
        v2f a0 = { d.x, d.y };
        v2f a1 = { d.z, d.w };
        acc = __builtin_amdgcn_wmma_f32_16x16x4_f32(
            false, a0, false, a0, (short)0, acc, false, false);
        acc = __builtin_amdgcn_wmma_f32_16x16x4_f32(
            false, a1, false, a1, (short)0, acc, false, false);
    }

    // D layout (32-bit C/D 16x16): lane group hi, reg v holds
    // D[M = v + 8*hi][N = r16]. Keep only the two diagonal 8x8 blocks.
    const bool valid = (hi == 0) ? (r16 < 8) : (r16 >= 8);
    if (valid) {
        const int nb  = pair * 2 + hi;    // which batch
        const int col = r16 & 7;
        float* base = ws + (size_t)nb * 64 + col;
        #pragma unroll
        for (int v = 0; v < 8; ++v) {
            atomicAdd(base + v * 8, acc[v]);   // global_atomic_add_f32
        }
    }
}

// ---------------------------------------------------------------------------
// Kernel 2: finalize. One thread per batch: window-average the 8x8 Gram into
// a 5x5 R, take the 15 triu entries, append 15 zeros (imag part), normalize.
// ---------------------------------------------------------------------------
__global__ void ssf_finalize(const float* __restrict__ ws,
                             float* __restrict__ out) {
    int n = blockIdx.x * blockDim.x + threadIdx.x;
    if (n >= N_BATCH) return;

    const float* C = ws + (size_t)n * 64;
    const float scale = 0.25f / (float)T_LEN;   // mean over J=4 windows, /T

    float R[5][5];
    #pragma unroll
    for (int a = 0; a < 5; ++a) {
        #pragma unroll
        for (int b = 0; b < 5; ++b) {
            float s = 0.0f;
            #pragma unroll
            for (int j = 0; j < 4; ++j)
                s += C[(j + a) * 8 + (j + b)];
            R[a][b] = s * scale;
        }
    }

    float tri[15];
    int idx = 0;
    #pragma unroll
    for (int a = 0; a < 5; ++a)
        #pragma unroll
        for (int b = a; b < 5; ++b)
            tri[idx++] = R[a][b];

    // feat = [tri(15), zeros(15)]; mean/std over all 30 entries.
    float sum = 0.0f;
    #pragma unroll
    for (int i = 0; i < 15; ++i) sum += tri[i];
    const float mean = sum * (1.0f / 30.0f);

    float var = 0.0f;
    #pragma unroll
    for (int i = 0; i < 15; ++i) {
        float d = tri[i] - mean;
        var += d * d;
    }
    var += 15.0f * mean * mean;                 // the 15 zero entries
    var *= (1.0f / 30.0f);

    const float inv = 1.0f / (sqrtf(var) + 1e-8f);

    float* o = out + (size_t)n * 30;
    #pragma unroll
    for (int i = 0; i < 15; ++i) o[i] = (tri[i] - mean) * inv;
    const float z = (0.0f - mean) * inv;
    #pragma unroll
    for (int i = 0; i < 15; ++i) o[15 + i] = z;
}

// ---------------------------------------------------------------------------
extern "C" void kernel_launch(void* const* d_in, const int* in_sizes, int n_in,
                              void* d_out, int out_size, void* d_ws, size_t ws_size,
                              hipStream_t stream) {
    const float* X   = (const float*)d_in[0];
    float*       out = (float*)d_out;
    float*       ws  = (float*)d_ws;            // needs 256*64*4 = 64 KB

    // 1) zero Gram accumulators
    const int ws_elems = N_BATCH * 64;
    ssf_zero_ws<<<(ws_elems + 255) / 256, 256, 0, stream>>>(ws, ws_elems);

    // 2) WMMA Gram accumulation: 128 pairs * 32 splits = 4096 waves
    const int total_waves = PAIRS * SPLITS;
    const int blocks = total_waves / WAVES_PER_BLOCK;   // 512
    ssf_gram_wmma<<<blocks, WAVES_PER_BLOCK * 32, 0, stream>>>(X, ws);

    // 3) finalize: one thread per batch
    ssf_finalize<<<1, N_BATCH, 0, stream>>>(ws, out);
}